// Classifer_38087769981032
// MI455X (gfx1250) — compile-verified
//
#include <hip/hip_runtime.h>
#include <hip/hip_bf16.h>
#include <math.h>

// ---------------------------------------------------------------------------
// Problem constants (match reference)
// ---------------------------------------------------------------------------
#define NN    100000   // nodes
#define NE    1600000  // edges
#define NG    2048     // graphs
#define D0    78       // input feature dim
#define D0P   80       // padded to multiple of 4 for WMMA K
#define DH    128      // hidden dim
#define BN_EPS 1e-5f

#define KC    128      // K-chunk staged in LDS
#define KCP   (KC + 2) // +2 floats bank padding (keeps b64 reads 8B-aligned)

typedef __attribute__((ext_vector_type(2))) float v2f;
typedef __attribute__((ext_vector_type(8))) float v8f;

// ---------------------------------------------------------------------------
// LDS-staged fp32 WMMA GEMM:  Z[M,Ncols] = A[M,K] @ B[K,Ncols] + bs*bias
// Block = 256 threads = 8 waves; block computes a 16-row x 128-col slab.
// Wave w owns column tile (blockIdx.y*8 + w).  K is processed in chunks of
// 128: A tile (16 x kc) and transposed B slab (128 x kc) staged in LDS,
// then the hot loop is 2x ds_load_b64 + 1x v_wmma_f32_16x16x4_f32 per step.
// grid = (M/16, ceil(Ncols/128)).  M mult of 16, K mult of 4.
// ---------------------------------------------------------------------------
__global__ __launch_bounds__(256) void gemm_f32_wmma(
    const float* __restrict__ A, const float* __restrict__ B,
    const float* __restrict__ bias, float bias_scale,
    float* __restrict__ Z, int K, int Ncols, int ntiles)
{
    __shared__ float sA[16 * KCP];    //  8.3 KB
    __shared__ float sB[128 * KCP];   // 66.6 KB (transposed: sB[col][k])

    const int tid   = threadIdx.x;
    const int lane  = tid & 31;
    const int wave  = tid >> 5;
    const int tile  = blockIdx.y * 8 + wave;    // global col tile (uniform/wave)
    const bool active = (tile < ntiles);
    const int row0  = blockIdx.x << 4;
    const int cbase = blockIdx.y << 7;          // first column of slab
    const int mlane = lane & 15;
    const int khalf = (lane >> 4) << 1;         // 0 or 2
    const int nloc  = (wave << 4) + mlane;      // local column 0..127

    // cooperative-load thread mapping
    const int ar = tid >> 4;                    // A row 0..15
    const int aj = tid & 15;                    // A col group
    const int bc = tid & 127;                   // B column 0..127
    const int bh = tid >> 7;                    // 0/1

    v8f c = {0.f, 0.f, 0.f, 0.f, 0.f, 0.f, 0.f, 0.f};

    for (int k0 = 0; k0 < K; k0 += KC) {
        int kc = K - k0; if (kc > KC) kc = KC;

        // stage A tile: sA[r][k] = A[(row0+r)*K + k0 + k]
        const float* __restrict__ ap = A + (long)(row0 + ar) * K + k0;
        for (int kk = aj; kk < kc; kk += 16)
            sA[ar * KCP + kk] = ap[kk];

        // stage B slab transposed: sB[c][k] = B[(k0+k)*Ncols + cbase + c]
        for (int kk = bh; kk < kc; kk += 2) {
            float v = (cbase + bc < Ncols)
                        ? B[(long)(k0 + kk) * Ncols + cbase + bc] : 0.f;
            sB[bc * KCP + kk] = v;
        }
        __syncthreads();

        if (active) {
            const float* __restrict__ sa = sA + mlane * KCP + khalf;
            const float* __restrict__ sb = sB + nloc  * KCP + khalf;
#pragma unroll 4
            for (int kk = 0; kk < kc; kk += 4) {
                float2 av = *reinterpret_cast<const float2*>(sa + kk); // ds_load_b64
                float2 bv = *reinterpret_cast<const float2*>(sb + kk); // ds_load_b64
                v2f a; a.x = av.x; a.y = av.y;
                v2f b; b.x = bv.x; b.y = bv.y;
                c = __builtin_amdgcn_wmma_f32_16x16x4_f32(
                        false, a, false, b, (short)0, c, false, false);
            }
        }
        __syncthreads();
    }

    if (!active) return;
    const int n0 = tile << 4;
    const float bv = bias_scale * bias[n0 + mlane];
    const int rbase = row0 + ((lane >> 4) << 3);
    float* __restrict__ zp = Z + (long)rbase * Ncols + n0 + mlane;
#pragma unroll
    for (int r = 0; r < 8; ++r)
        zp[(long)r * Ncols] = c[r] + bv;
}

// ---------------------------------------------------------------------------
// agg[i] = 2 * x[i]
// ---------------------------------------------------------------------------
__global__ void scale2_kernel(const float* __restrict__ s,
                              float* __restrict__ d, long n)
{
    long i = (long)blockIdx.x * blockDim.x + threadIdx.x;
    if (i < n) d[i] = 2.0f * s[i];
}

// ---------------------------------------------------------------------------
// agg[dst[e], :] += x[src[e], :]   (4 floats per thread, hw f32 atomics)
// grid = (ceil(E/256), ld/4)
// ---------------------------------------------------------------------------
__global__ void scatter_edges(const float* __restrict__ x,
                              const int* __restrict__ src,
                              const int* __restrict__ dst,
                              float* __restrict__ agg, int E, int ld)
{
    int e = blockIdx.x * blockDim.x + threadIdx.x;
    if (e >= E) return;
    int c4 = blockIdx.y << 2;
    const float4 v = *reinterpret_cast<const float4*>(
        x + (long)src[e] * ld + c4);
    float* p = agg + (long)dst[e] * ld + c4;
    unsafeAtomicAdd(p + 0, v.x);
    unsafeAtomicAdd(p + 1, v.y);
    unsafeAtomicAdd(p + 2, v.z);
    unsafeAtomicAdd(p + 3, v.w);
}

// ---------------------------------------------------------------------------
// Column sum / sumsq reduction for BatchNorm. blockDim == Ncols.
// ---------------------------------------------------------------------------
__global__ void col_stats(const float* __restrict__ Z, float* __restrict__ st,
                          int M, int Ncols, int rpb)
{
    int c  = threadIdx.x;
    int r0 = blockIdx.x * rpb;
    int r1 = r0 + rpb; if (r1 > M) r1 = M;
    float s = 0.f, q = 0.f;
    for (int r = r0; r < r1; ++r) {
        float v = Z[(long)r * Ncols + c];
        s += v; q += v * v;
    }
    unsafeAtomicAdd(&st[c],         s);
    unsafeAtomicAdd(&st[Ncols + c], q);
}

// ---------------------------------------------------------------------------
// X = relu( (Z - mu) * rsqrt(var+eps) * gamma + beta ).  Ncols power of two.
// ---------------------------------------------------------------------------
__global__ void bn_relu(const float* __restrict__ Z,
                        const float* __restrict__ st,
                        const float* __restrict__ gamma,
                        const float* __restrict__ beta,
                        float* __restrict__ X,
                        long total, int cmask, int Ncols, float invM)
{
    long i = (long)blockIdx.x * blockDim.x + threadIdx.x;
    if (i >= total) return;
    int c = (int)i & cmask;
    float mu  = st[c] * invM;
    float var = st[Ncols + c] * invM - mu * mu;
    float v = (Z[i] - mu) * rsqrtf(var + BN_EPS) * gamma[c] + beta[c];
    X[i] = v > 0.f ? v : 0.f;
}

__global__ void zero_f32(float* __restrict__ p, long n)
{
    long i = (long)blockIdx.x * blockDim.x + threadIdx.x;
    if (i < n) p[i] = 0.f;
}

// ---------------------------------------------------------------------------
// hpad[n, 0:80] = { h[n, 0:78], 0, 0 }     grid = (ceil(N/256), 20)
// ---------------------------------------------------------------------------
__global__ void pad_h_kernel(const float* __restrict__ h,
                             float* __restrict__ hp, int Nn)
{
    int n = blockIdx.x * blockDim.x + threadIdx.x;
    if (n >= Nn) return;
    int c = blockIdx.y << 2;
    float4 v;
    v.x = (c + 0 < D0) ? h[(long)n * D0 + c + 0] : 0.f;
    v.y = (c + 1 < D0) ? h[(long)n * D0 + c + 1] : 0.f;
    v.z = (c + 2 < D0) ? h[(long)n * D0 + c + 2] : 0.f;
    v.w = (c + 3 < D0) ? h[(long)n * D0 + c + 3] : 0.f;
    *reinterpret_cast<float4*>(hp + (long)n * D0P + c) = v;
}

// W1pad[80,128] from W1[78,128]
__global__ void pad_w1_kernel(const float* __restrict__ W,
                              float* __restrict__ Wp)
{
    int i = blockIdx.x * blockDim.x + threadIdx.x;   // 80*128
    if (i >= D0P * DH) return;
    int k = i >> 7, c = i & 127;
    Wp[i] = (k < D0) ? W[k * DH + c] : 0.f;
}

// xg[gid[n], :] += x[n, :]    grid = (ceil(N/256), 32)
__global__ void readout_scatter(const float* __restrict__ x,
                                const int* __restrict__ gid,
                                float* __restrict__ xg, int Nn)
{
    int n = blockIdx.x * blockDim.x + threadIdx.x;
    if (n >= Nn) return;
    int c4 = blockIdx.y << 2;
    int g  = gid[n];
    const float4 v = *reinterpret_cast<const float4*>(x + (long)n * DH + c4);
    float* p = xg + (long)g * DH + c4;
    unsafeAtomicAdd(p + 0, v.x);
    unsafeAtomicAdd(p + 1, v.y);
    unsafeAtomicAdd(p + 2, v.z);
    unsafeAtomicAdd(p + 3, v.w);
}

// Wp[256,208] = fc2_W[:, 204:408] zero-padded; bp likewise.
__global__ void prepack_fc2(const float* __restrict__ W,
                            const float* __restrict__ b,
                            float* __restrict__ Wp, float* __restrict__ bp)
{
    int k = blockIdx.x;          // 0..255
    int j = threadIdx.x;         // 0..207
    if (j >= 208) return;
    Wp[k * 208 + j] = (j < 204) ? W[k * 408 + 204 + j] : 0.f;
    if (k == 0) bp[j] = (j < 204) ? b[204 + j] : 0.f;
}

// out[g, 0:204] = sigmoid(logits[g, 0:204])   grid = 2048, block = 256
__global__ void sigmoid_tail(const float* __restrict__ logits,
                             float* __restrict__ out)
{
    int g = blockIdx.x, j = threadIdx.x;
    if (j >= 204) return;
    float v = logits[(long)g * 208 + j];
    out[(long)g * 204 + j] = 1.0f / (1.0f + expf(-v));
}

// ---------------------------------------------------------------------------
// Host-side orchestration
// ---------------------------------------------------------------------------
static inline int ceil_div(long a, long b) { return (int)((a + b - 1) / b); }

extern "C" void kernel_launch(void* const* d_in, const int* in_sizes, int n_in,
                              void* d_out, int out_size, void* d_ws, size_t ws_size,
                              hipStream_t stream)
{
    const float* h        = (const float*)d_in[0];
    const int*   esrc     = (const int*)  d_in[1];
    const int*   edst     = (const int*)  d_in[2];
    const int*   gid      = (const int*)  d_in[3];
    const float* W1       = (const float*)d_in[4];
    const float* b1       = (const float*)d_in[5];
    const float* g1_gamma = (const float*)d_in[6];
    const float* g1_beta  = (const float*)d_in[7];
    const float* Wg       = (const float*)d_in[8];
    const float* bg       = (const float*)d_in[9];
    const float* g_gamma  = (const float*)d_in[10];
    const float* g_beta   = (const float*)d_in[11];
    const float* fc1_W    = (const float*)d_in[12];
    const float* fc1_b    = (const float*)d_in[13];
    const float* bn1_g    = (const float*)d_in[14];
    const float* bn1_b    = (const float*)d_in[15];
    const float* lin_W    = (const float*)d_in[16];
    const float* lin_b    = (const float*)d_in[17];
    const float* lbn_g    = (const float*)d_in[18];
    const float* lbn_b    = (const float*)d_in[19];
    const float* fc2_W    = (const float*)d_in[20];
    const float* fc2_b    = (const float*)d_in[21];
    float* out = (float*)d_out;
    float* ws  = (float*)d_ws;

    // workspace layout (floats, all offsets multiples of 64)
    float* HPAD  = ws;                        // N*80
    float* X     = HPAD + (long)NN * D0P;     // N*128 activations
    float* AGG   = X    + (long)NN * DH;      // N*128 (also N*80 for layer 0)
    float* ZB    = AGG  + (long)NN * DH;      // N*128 pre-BN, reused by head
    float* W1P   = ZB   + (long)NN * DH;      // 80*128
    float* STATS = W1P  + (long)D0P * DH;     // 1024
    float* XG    = STATS + 1024;              // 2048*128
    float* WP    = XG   + (long)NG * DH;      // 256*208
    float* BP    = WP   + 256L * 208;         // 256

    const dim3 blk(256);

    // ---- layer-0 input packing --------------------------------------------
    pad_h_kernel<<<dim3(ceil_div(NN, 256), D0P / 4), blk, 0, stream>>>(h, HPAD, NN);
    pad_w1_kernel<<<ceil_div((long)D0P * DH, 256), blk, 0, stream>>>(W1, W1P);

    // ---- 5 GIN layers ------------------------------------------------------
    for (int layer = 0; layer < 5; ++layer) {
        const float* xin  = (layer == 0) ? HPAD : X;
        const int    ld   = (layer == 0) ? D0P  : DH;
        const float* Wl   = (layer == 0) ? W1P  : (Wg + (long)(layer - 1) * DH * DH);
        const float* bl   = (layer == 0) ? b1   : (bg + (long)(layer - 1) * DH);
        const float* gam  = (layer == 0) ? g1_gamma : (g_gamma + (long)(layer - 1) * DH);
        const float* bet  = (layer == 0) ? g1_beta  : (g_beta  + (long)(layer - 1) * DH);

        zero_f32<<<1, 256, 0, stream>>>(STATS, 2L * DH);
        scale2_kernel<<<ceil_div((long)NN * ld, 256), blk, 0, stream>>>(
            xin, AGG, (long)NN * ld);
        scatter_edges<<<dim3(ceil_div(NE, 256), ld / 4), blk, 0, stream>>>(
            xin, esrc, edst, AGG, NE, ld);
        gemm_f32_wmma<<<dim3(NN / 16, 1), blk, 0, stream>>>(
            AGG, Wl, bl, 2.0f, ZB, ld, DH, DH / 16);
        col_stats<<<200, DH, 0, stream>>>(ZB, STATS, NN, DH, 500);
        bn_relu<<<ceil_div((long)NN * DH, 256), blk, 0, stream>>>(
            ZB, STATS, gam, bet, X, (long)NN * DH, DH - 1, DH, 1.0f / NN);
    }

    // ---- graph readout -----------------------------------------------------
    zero_f32<<<ceil_div((long)NG * DH, 256), blk, 0, stream>>>(XG, (long)NG * DH);
    readout_scatter<<<dim3(ceil_div(NN, 256), DH / 4), blk, 0, stream>>>(X, gid, XG, NN);

    // ---- fc1: [2048,128] @ [128,512] + BN + relu --------------------------
    zero_f32<<<1, 256, 0, stream>>>(STATS, 1024L);
    gemm_f32_wmma<<<dim3(NG / 16, 4), blk, 0, stream>>>(
        XG, fc1_W, fc1_b, 1.0f, ZB, 128, 512, 32);
    col_stats<<<16, 512, 0, stream>>>(ZB, STATS, NG, 512, 128);
    bn_relu<<<ceil_div((long)NG * 512, 256), blk, 0, stream>>>(
        ZB, STATS, bn1_g, bn1_b, AGG, (long)NG * 512, 511, 512, 1.0f / NG);

    // ---- lin: [2048,512] @ [512,256] + BN + relu --------------------------
    zero_f32<<<1, 256, 0, stream>>>(STATS, 512L);
    gemm_f32_wmma<<<dim3(NG / 16, 2), blk, 0, stream>>>(
        AGG, lin_W, lin_b, 1.0f, ZB, 512, 256, 16);
    col_stats<<<16, 256, 0, stream>>>(ZB, STATS, NG, 256, 128);
    bn_relu<<<ceil_div((long)NG * 256, 256), blk, 0, stream>>>(
        ZB, STATS, lbn_g, lbn_b, X, (long)NG * 256, 255, 256, 1.0f / NG);

    // ---- fc2 (only last 204 columns) + sigmoid ----------------------------
    prepack_fc2<<<256, 208, 0, stream>>>(fc2_W, fc2_b, WP, BP);
    gemm_f32_wmma<<<dim3(NG / 16, 2), blk, 0, stream>>>(
        X, WP, BP, 1.0f, ZB, 256, 208, 13);
    sigmoid_tail<<<NG, 256, 0, stream>>>(ZB, out);
}